// ActionNetwork_20401094656134
// MI455X (gfx1250) — compile-verified
//
#include <hip/hip_runtime.h>

typedef __attribute__((ext_vector_type(2))) float v2f;
typedef __attribute__((ext_vector_type(8))) float v8f;

static inline int cdiv(int a, int b) { return (a + b - 1) / b; }

// ---------------------------------------------------------------- degree prep
__global__ void k_fill1(float* __restrict__ p, int n) {
  int i = blockIdx.x * blockDim.x + threadIdx.x;
  if (i < n) p[i] = 1.0f;                       // +1 from added self loop
}

__global__ void k_degree(const int* __restrict__ src, const int* __restrict__ dst,
                         float* __restrict__ deg, int E) {
  int e = blockIdx.x * blockDim.x + threadIdx.x;
  if (e < E) {
    int s = src[e], d = dst[e];
    if (s != d) atomicAdd(&deg[d], 1.0f);       // w = (src != dst)
  }
}

__global__ void k_rsqrt(const float* __restrict__ deg, float* __restrict__ dis,
                        float* __restrict__ dis2, int n) {
  int i = blockIdx.x * blockDim.x + threadIdx.x;
  if (i < n) {
    float dg = deg[i];
    dis[i]  = rsqrtf(dg);
    dis2[i] = 1.0f / dg;                        // self-loop coefficient dis*dis
  }
}

__global__ void k_norm(const int* __restrict__ src, const int* __restrict__ dst,
                       const float* __restrict__ dis, float* __restrict__ nrm, int E) {
  int e = blockIdx.x * blockDim.x + threadIdx.x;
  if (e < E) {
    int s = src[e], d = dst[e];
    nrm[e] = (s != d) ? dis[s] * dis[d] : 0.0f;
  }
}

// ------------------------------------------------------------- WMMA GEMM (fused)
//   xw[n,NC] = relu?(X[n,64]) @ W[64,NC]
//   h [n,NC] = dis2[row] * xw + bias[col]        (aggregation self-loop + bias init)
// fp32 WMMA 16x16x4, wave32, one wave per 16x16 output tile.
// A (16x4 f32):  lane row = lane&15, K pair = (lane>>4)*2  -> contiguous v2f load.
// B (4x16 f32):  lane col = lane&15, K pair = (lane>>4)*2  (all 16 frags preloaded).
// C/D (16x16):   VGPR j -> M=j (lanes 0-15), M=j+8 (lanes 16-31).
// NCOLS/RELU are compile-time: all index math is 32-bit shifts (no v_mul_u64),
// no cndmask-select for the relu path. Requires n % 16 == 0 so the row guard is
// wave-uniform (EXEC stays all-1s for WMMA).
template <int NCOLS, bool RELU>
__global__ void k_gemm64_wmma(const float* __restrict__ X, const float* __restrict__ W,
                              const float* __restrict__ dis2, const float* __restrict__ bias,
                              float* __restrict__ xw, float* __restrict__ h, int n) {
  constexpr int NTILES = (NCOLS + 15) / 16;    // 4 (NC=64) or 1 (NC=4)
  constexpr int MTPB   = 8 / NTILES;           // m-tiles per 8-wave block
  const int lane  = threadIdx.x & 31;
  const int wave  = threadIdx.x >> 5;
  const int mtile = wave / NTILES;
  const int ntile = wave % NTILES;
  const int row0  = (blockIdx.x * MTPB + mtile) * 16;
  if (row0 >= n) return;                       // wave-uniform: tiles are always full

  const int kb   = (lane >> 4) << 1;           // 0 or 2
  const int bcol = ntile * 16 + (lane & 15);
  const int bc   = (NCOLS >= 16) ? bcol
                                 : ((bcol < NCOLS) ? bcol : (NCOLS - 1)); // clamp
  const unsigned aoff = (unsigned)(row0 + (lane & 15)) * 64u + (unsigned)kb;

  // Preload all B fragments: W is tiny (<=16KB) and grid-invariant -> L0/L2 hit.
  v2f bfrag[16];
  #pragma unroll
  for (int t = 0; t < 16; ++t) {
    bfrag[t].x = W[(unsigned)(4 * t + kb)     * NCOLS + bc];
    bfrag[t].y = W[(unsigned)(4 * t + kb + 1) * NCOLS + bc];
  }

  v8f c = {};
  #pragma unroll
  for (int t = 0; t < 16; ++t) {
    v2f a = *(const v2f*)(X + aoff + 4 * t);
    if (RELU) { a.x = fmaxf(a.x, 0.0f); a.y = fmaxf(a.y, 0.0f); }
    c = __builtin_amdgcn_wmma_f32_16x16x4_f32(false, a, false, bfrag[t],
                                              (short)0, c, false, false);
  }

  if (bcol < NCOLS) {                          // single tail branch, after the loop
    const int   mofs = (lane >> 4) << 3;       // 0 or 8
    const float bv   = bias[bcol];
    #pragma unroll
    for (int j = 0; j < 8; ++j) {
      const int      r = row0 + mofs + j;
      const unsigned o = (unsigned)r * NCOLS + (unsigned)bcol;
      const float    v = c[j];
      xw[o] = v;                               // gather source
      h [o] = dis2[r] * v + bv;                // self-loop + bias init
    }
  }
}

// --------------------------- scatter width 64: out[dst,:] += norm * xw[src,:]
// one thread = one float4 chunk of one edge  (E*16 threads)
__global__ void k_scatter64(const int* __restrict__ src, const int* __restrict__ dst,
                            const float* __restrict__ nrm, const float* __restrict__ xw,
                            float* __restrict__ out, int E) {
  int idx = blockIdx.x * blockDim.x + threadIdx.x;
  if (idx >= E * 16) return;
  int e = idx >> 4;
  unsigned q = ((unsigned)idx & 15u) << 2;
  float nv = nrm[e];
  if (nv == 0.0f) return;                    // self-loop edges contribute nothing
  const float4 v = *(const float4*)(xw + (unsigned)src[e] * 64u + q);
  float* op = out + (unsigned)dst[e] * 64u + q;
  atomicAdd(op + 0, nv * v.x);
  atomicAdd(op + 1, nv * v.y);
  atomicAdd(op + 2, nv * v.z);
  atomicAdd(op + 3, nv * v.w);
}

// ---------------------------- scatter width 4 (projected layer-3 aggregation)
__global__ void k_scatter4(const int* __restrict__ src, const int* __restrict__ dst,
                           const float* __restrict__ nrm, const float* __restrict__ xw,
                           float* __restrict__ out, int E) {
  int e = blockIdx.x * blockDim.x + threadIdx.x;
  if (e >= E) return;
  float nv = nrm[e];
  if (nv == 0.0f) return;
  const float4 v = *(const float4*)(xw + (unsigned)src[e] * 4u);
  float* op = out + (unsigned)dst[e] * 4u;
  atomicAdd(op + 0, nv * v.x);
  atomicAdd(op + 1, nv * v.y);
  atomicAdd(op + 2, nv * v.z);
  atomicAdd(op + 3, nv * v.w);
}

// ---------------------------------------------------------------------- launch
extern "C" void kernel_launch(void* const* d_in, const int* in_sizes, int n_in,
                              void* d_out, int out_size, void* d_ws, size_t ws_size,
                              hipStream_t stream) {
  const float* x  = (const float*)d_in[0];
  const int*   ei = (const int*)d_in[1];
  const float* W0 = (const float*)d_in[2];
  const float* b0 = (const float*)d_in[3];
  const float* W1 = (const float*)d_in[4];
  const float* b1 = (const float*)d_in[5];
  const float* W2 = (const float*)d_in[6];
  const float* b2 = (const float*)d_in[7];

  const int N = in_sizes[0] / 64;            // 100000
  const int E = in_sizes[1] / 2;             // 1600000
  const int* src = ei;                       // edge_index[0]
  const int* dst = ei + E;                   // edge_index[1]

  float* ws   = (float*)d_ws;
  float* deg  = ws;                          // N
  float* dis  = deg  + N;                    // N
  float* dis2 = dis  + N;                    // N
  float* nrm  = dis2 + N;                    // E
  float* xw   = nrm  + E;                    // N*64 (holds N*4 in layer 3)
  float* h    = xw   + (size_t)N * 64;       // N*64
  float* out  = (float*)d_out;               // N*4

  const int T = 256;

  // ---- graph normalization (computed once, shared by all 3 layers)
  k_fill1 <<<cdiv(N, T), T, 0, stream>>>(deg, N);
  k_degree<<<cdiv(E, T), T, 0, stream>>>(src, dst, deg, E);
  k_rsqrt <<<cdiv(N, T), T, 0, stream>>>(deg, dis, dis2, N);
  k_norm  <<<cdiv(E, T), T, 0, stream>>>(src, dst, dis, nrm, E);

  // ---- layer 0: xw = x@W0 ; h = dis2*xw + b0 ; h[dst] += nrm*xw[src]
  k_gemm64_wmma<64, false><<<cdiv(N, 32), T, 0, stream>>>(x, W0, dis2, b0, xw, h, N);
  k_scatter64<<<cdiv(E * 16, T), T, 0, stream>>>(src, dst, nrm, xw, h, E);

  // ---- layer 1 (ReLU fused into A-matrix load)
  k_gemm64_wmma<64, true><<<cdiv(N, 32), T, 0, stream>>>(h, W1, dis2, b1, xw, h, N);
  k_scatter64<<<cdiv(E * 16, T), T, 0, stream>>>(src, dst, nrm, xw, h, E);

  // ---- layer 2: project to width 4 BEFORE aggregation (16x less scatter traffic)
  k_gemm64_wmma<4, true><<<cdiv(N, 128), T, 0, stream>>>(h, W2, dis2, b2, xw, out, N);
  k_scatter4<<<cdiv(E, T), T, 0, stream>>>(src, dst, nrm, xw, out, E);
}